// AdaptiveRenderer_41549513622103
// MI455X (gfx1250) — compile-verified
//
#include <hip/hip_runtime.h>
#include <hip/hip_bf16.h>

// Problem geometry (from reference)
#define KS    21
#define CROP  10
#define HH    192
#define WW    192
#define OHH   172            // 192 - 21 + 1
#define OWW   172
#define NB    4
#define NC    3
#define TAPS  (KS * KS)      // 441
#define PLANE (HH * WW)      // 36864 elements per [b,j] plane

// Tiling: 64x4 output pixels per block, 2 pixels per thread.
// Block = 32x4 = 128 threads = 4 wave32s; each wave row does 64 contiguous pixels.
#define TW   64                      // tile width in pixels
#define TH   4                       // tile height
#define NTX  (TW / 2)                // 32 threads across
#define NTHREADS (NTX * TH)          // 128
#define PW   (TW + KS - 1)           // 84
#define PH   (TH + KS - 1)           // 24

#define LOG2E 1.4426950408889634f
#define LN2   0.6931471805599453f

// Native clang vector type: accepted by __builtin_nontemporal_{load,store},
// and eligible for V_PK_*_F32 packed-math lowering.
typedef __attribute__((ext_vector_type(2))) float vf2;

// CDNA5 async global->LDS b32 copy (tracked by ASYNCcnt).
__device__ __forceinline__ void async_g2lds_b32(unsigned lds_addr, const float* gptr) {
    asm volatile("global_load_async_to_lds_b32 %0, %1, off"
                 :: "v"(lds_addr), "v"(gptr)
                 : "memory");
}

__global__ __launch_bounds__(NTHREADS)
void AdaptiveRenderer_kpn_kernel(const float* __restrict__ kraw,   // [B,441,192,192]
                                 const float* __restrict__ rad,    // [B,3,192,192]
                                 const float* __restrict__ mask,   // [B,1,192,192]
                                 float* __restrict__ wsum,         // [B,3,172,172]
                                 float* __restrict__ ksum)         // [B,172,172]
{
    // Interleaved {r0,r1,r2,mask} halo patch -> ds_load_b128 per patch element.
    __shared__ float4 patch[PH * PW];                              // 32256 B

    const int b   = blockIdx.z;
    const int oy0 = blockIdx.y * TH;                               // < 172 always (43*4)
    const int ox0 = blockIdx.x * TW;                               // may exceed 172 on last tile
    const int tx  = threadIdx.x;                                   // 0..31
    const int ty  = threadIdx.y;                                   // 0..3
    const int tid = ty * NTX + tx;

    // ---- Stage radiance(3ch)+mask halo tile into LDS via async-tensor path ----
    const float* r0 = rad  + (size_t)(b * NC + 0) * PLANE;
    const float* r1 = rad  + (size_t)(b * NC + 1) * PLANE;
    const float* r2 = rad  + (size_t)(b * NC + 2) * PLANE;
    const float* mp = mask + (size_t)b * PLANE;

    for (int i = tid; i < PH * PW; i += NTHREADS) {
        const int py = i / PW;
        const int px = i - py * PW;
        int gy = oy0 + py; if (gy > HH - 1) gy = HH - 1;           // clamp (affects only
        int gx = ox0 + px; if (gx > WW - 1) gx = WW - 1;           // never-stored pixels)
        const int g = gy * WW + gx;
        const unsigned lds = (unsigned)(size_t)&patch[i];          // low 32 bits = LDS offset
        async_g2lds_b32(lds      , r0 + g);
        async_g2lds_b32(lds +  4u, r1 + g);
        async_g2lds_b32(lds +  8u, r2 + g);
        async_g2lds_b32(lds + 12u, mp + g);
    }
    asm volatile("s_wait_asynccnt 0x0" ::: "memory");              // drain ASYNCcnt
    __syncthreads();

    // ---- Two horizontally adjacent output pixels per thread ----
    const int oh  = oy0 + ty;                                      // < 172 always
    const int ow0 = ox0 + 2 * tx;                                  // even
    // Clamp so the b64 load reads two valid columns (owe, owe+1 both <= 171).
    const int owe = (ow0 < OWW - 1) ? ow0 : (OWW - 2);             // even -> 8B-aligned loads

    const float* kdy = kraw + (size_t)b * TAPS * PLANE
                            + (size_t)(oh + CROP) * WW + (owe + CROP);
    const float* kpf = kdy + (size_t)(2 * KS) * PLANE;             // prefetch 42 planes ahead

    vf2   rgA = {0.f, 0.f}, rgC = {0.f, 0.f};                      // {r,g} accumulators (packed)
    float bA = 0.f, kA = 0.f, bC = 0.f, kC = 0.f;                  // b-channel + kernel_sum
    const float4* prow = &patch[ty * PW + 2 * tx];

    for (int dy = 0; dy < KS; ++dy) {
        // Branch-free prefetch of one full future tap-row (imm offsets off pfb).
        // Tail rows (dy >= 19) harmlessly re-prefetch the current (hot) row.
        const float* pfb = (dy < KS - 2) ? kpf : kdy;
#pragma unroll
        for (int dx = 0; dx < KS; ++dx)
            __builtin_prefetch(pfb + dx * PLANE, 0, 3);
        kpf += KS * PLANE;

        const float4* p  = prow + dy * PW;
        const vf2*    k2 = (const vf2*)kdy;                        // 8B-aligned (even offsets)
#pragma unroll
        for (int dx = 0; dx < KS; ++dx) {
            // One b64 NT load = taps for both pixels (consume-once stream).
            const vf2    x2 = __builtin_nontemporal_load(k2 + dx * (PLANE / 2));
            const float4 t0 = p[dx];                               // ds_load_b128, imm offset
            const float4 t1 = p[dx + 1];                           // CSE'd with next iter's t0

            // softplus(x) = ln2 * log2(1 + 2^(x*log2e)); raw v_exp/v_log, no fixup.
            const vf2   ei  = x2 * LOG2E;                          // v_pk_mul_f32 (pair adjacent)
            const float e0  = __builtin_amdgcn_exp2f(ei.x);
            const float e1  = __builtin_amdgcn_exp2f(ei.y);
            const float sp0 = LN2 * __builtin_amdgcn_logf(1.0f + e0);
            const float sp1 = LN2 * __builtin_amdgcn_logf(1.0f + e1);

            const float wm0 = sp0 * t0.w;                          // kernel * mask
            const float wm1 = sp1 * t1.w;

            vf2 rg0; rg0.x = t0.x; rg0.y = t0.y;                   // adjacent regs from b128
            vf2 rg1; rg1.x = t1.x; rg1.y = t1.y;
            rgA += rg0 * wm0;                                      // v_pk_fma_f32 (splat wm)
            rgC += rg1 * wm1;
            bA = fmaf(wm0, t0.z, bA);
            bC = fmaf(wm1, t1.z, bC);
            kA += wm0;
            kC += wm1;
        }
        kdy += KS * PLANE;                                         // one addr update per 21 taps
    }

    // ow0 even, OWW even: ow0 < OWW implies ow0+1 <= OWW-1, one guard covers the pair.
    if (ow0 < OWW) {
        const size_t o    = (size_t)oh * OWW + ow0;                // even -> 8B-aligned stores
        const size_t chpl = (size_t)OHH * OWW;
        vf2 s0; s0.x = rgA.x; s0.y = rgC.x;                        // ch0 for pixel pair
        vf2 s1; s1.x = rgA.y; s1.y = rgC.y;                        // ch1
        vf2 s2; s2.x = bA;    s2.y = bC;                           // ch2
        vf2 sk; sk.x = kA;    sk.y = kC;                           // kernel_sum
        __builtin_nontemporal_store(s0, (vf2*)&wsum[((size_t)b * NC + 0) * chpl + o]);
        __builtin_nontemporal_store(s1, (vf2*)&wsum[((size_t)b * NC + 1) * chpl + o]);
        __builtin_nontemporal_store(s2, (vf2*)&wsum[((size_t)b * NC + 2) * chpl + o]);
        __builtin_nontemporal_store(sk, (vf2*)&ksum[(size_t)b * chpl + o]);
    }
}

extern "C" void kernel_launch(void* const* d_in, const int* in_sizes, int n_in,
                              void* d_out, int out_size, void* d_ws, size_t ws_size,
                              hipStream_t stream) {
    const float* kraw = (const float*)d_in[0];   // [4,441,192,192]
    const float* rad  = (const float*)d_in[1];   // [4,3,192,192]
    const float* mask = (const float*)d_in[2];   // [4,1,192,192]

    float* wsum = (float*)d_out;                                   // [4,3,172,172]
    float* ksum = wsum + (size_t)NB * NC * OHH * OWW;              // [4,172,172]

    dim3 block(NTX, TH, 1);                                        // 128 = 4 wave32s
    dim3 grid((OWW + TW - 1) / TW,                                 // 3
              OHH / TH,                                            // 43 (exact)
              NB);                                                 // 4
    AdaptiveRenderer_kpn_kernel<<<grid, block, 0, stream>>>(kraw, rad, mask, wsum, ksum);
}